// ASGNN_1614907703644
// MI455X (gfx1250) — compile-verified
//
#include <hip/hip_runtime.h>
#include <hip/hip_bf16.h>

// ASGNN on MI455X (gfx1250).
// - Attention branch is dead (softmax over size-1 axis == 1): Wal/bal/War unused.
// - h is never materialized: only t=h@W2l^T, r=h@W2r^T needed (mean-agg commutes
//   with the 256->1 linear), computed inside the conv1 WMMA kernel.
// - Graph aggregation is CSR-gather based: no float atomics anywhere.

#define NN 100000
#define NE 1600000
#define FF 128
#define HH 256

typedef __attribute__((ext_vector_type(2))) float v2f;
typedef __attribute__((ext_vector_type(8))) float v8f;

// ---------------- kernel 0: zero int counters ----------------
__global__ __launch_bounds__(256) void zero_int_kernel(int* __restrict__ p, int n) {
    int i = blockIdx.x * 256 + threadIdx.x;
    int stride = gridDim.x * 256;
    for (; i < n; i += stride) p[i] = 0;
}

// ---------------- kernel 1: in-degree counts (int) ----------------
__global__ __launch_bounds__(256) void count_kernel(const long long* __restrict__ ei,
                                                    int* __restrict__ cnt) {
    int e = blockIdx.x * 256 + threadIdx.x;
    if (e < NE) atomicAdd(&cnt[(int)ei[NE + e]], 1);
}

// ---------------- kernels 2a/2b/2c: exclusive scan of cnt -> offs ----------
// 2a: per-1024-element block scan (256 thr x 4 seq), records block totals
__global__ __launch_bounds__(256) void scan1_kernel(const int* __restrict__ cnt,
                                                    int* __restrict__ offs,
                                                    int* __restrict__ blksum) {
    __shared__ int sd[256];
    const int tid = threadIdx.x;
    const int base = blockIdx.x * 1024 + tid * 4;
    int v[4];
    int s = 0;
#pragma unroll
    for (int j = 0; j < 4; ++j) {
        int idx = base + j;
        int c = (idx < NN) ? cnt[idx] : 0;
        v[j] = s;          // exclusive within thread
        s += c;
    }
    sd[tid] = s;
    __syncthreads();
    for (int off = 1; off < 256; off <<= 1) {    // Hillis-Steele inclusive
        int t = (tid >= off) ? sd[tid - off] : 0;
        __syncthreads();
        sd[tid] += t;
        __syncthreads();
    }
    int excl = (tid > 0) ? sd[tid - 1] : 0;
#pragma unroll
    for (int j = 0; j < 4; ++j) {
        int idx = base + j;
        if (idx < NN) offs[idx] = excl + v[j];
    }
    if (tid == 255) blksum[blockIdx.x] = sd[255];
}

// 2b: scan the (<=128) block totals in-place -> exclusive block offsets
__global__ __launch_bounds__(128) void scan2_kernel(int* __restrict__ blksum, int nb) {
    __shared__ int sd[128];
    const int tid = threadIdx.x;
    sd[tid] = (tid < nb) ? blksum[tid] : 0;
    __syncthreads();
    for (int off = 1; off < 128; off <<= 1) {
        int t = (tid >= off) ? sd[tid - off] : 0;
        __syncthreads();
        sd[tid] += t;
        __syncthreads();
    }
    if (tid < nb) blksum[tid] = (tid > 0) ? sd[tid - 1] : 0;
}

// 2c: add block offsets
__global__ __launch_bounds__(256) void scan3_kernel(int* __restrict__ offs,
                                                    const int* __restrict__ blksum) {
    int i = blockIdx.x * 256 + threadIdx.x;
    if (i < NN) offs[i] += blksum[i >> 10];
}

// ---------------- kernel 3: fill CSR (group src by dst) ----------------
__global__ __launch_bounds__(256) void fill_kernel(const long long* __restrict__ ei,
                                                   const int* __restrict__ offs,
                                                   int* __restrict__ fill,
                                                   int* __restrict__ csr) {
    int e = blockIdx.x * 256 + threadIdx.x;
    if (e < NE) {
        int src = (int)ei[e];
        int dst = (int)ei[NE + e];
        int pos = atomicAdd(&fill[dst], 1);
        csr[offs[dst] + pos] = src;
    }
}

// ---------------- kernel 4: CSR gather-mean of x -> aggNorm --------------
// one wave per node; lane l owns features 4l..4l+3 in registers; no atomics.
__global__ __launch_bounds__(256) void aggregate_kernel(const float* __restrict__ x,
                                                        const int* __restrict__ csr,
                                                        const int* __restrict__ offs,
                                                        const int* __restrict__ cnt,
                                                        float* __restrict__ aggNorm) {
    int node = blockIdx.x * 8 + (threadIdx.x >> 5);
    if (node >= NN) return;
    const int lane = threadIdx.x & 31;
    const int beg = offs[node];
    const int deg = cnt[node];
    float4 acc = {0.0f, 0.0f, 0.0f, 0.0f};
    for (int j = 0; j < deg; ++j) {
        int s = csr[beg + j];                       // wave-uniform
        if (j + 1 < deg)                            // gfx1250 global_prefetch_b8
            __builtin_prefetch(x + (size_t)csr[beg + j + 1] * FF, 0, 0);
        const float4 v = *(const float4*)(x + (size_t)s * FF + lane * 4);
        acc.x += v.x; acc.y += v.y; acc.z += v.z; acc.w += v.w;
    }
    const float inv = 1.0f / fmaxf((float)deg, 1.0f);
    float4 o;
    o.x = acc.x * inv; o.y = acc.y * inv; o.z = acc.z * inv; o.w = acc.w * inv;
    *(float4*)(aggNorm + (size_t)node * FF + lane * 4) = o;
}

// ---------------- kernel 5: conv1 (WMMA f32) + fused ReLU + W2 dots ---------
// Block = 256 threads (8 waves), one 16-row node tile. Tiles staged into LDS
// with GLOBAL_LOAD_ASYNC_TO_LDS_B128 (ASYNCcnt). h tile lives only in LDS and
// is reduced to t[row]=h.W2l, r[row]=h.W2r.
__global__ __launch_bounds__(256) void conv1_kernel(const float* __restrict__ x,
                                                    const float* __restrict__ aggN,
                                                    const float* __restrict__ W1l,
                                                    const float* __restrict__ b1l,
                                                    const float* __restrict__ W1r,
                                                    const float* __restrict__ W2l,
                                                    const float* __restrict__ W2r,
                                                    float* __restrict__ t_out,
                                                    float* __restrict__ r_out) {
    // pads: 132 -> conflict-free A-frag column reads; 260 -> conflict-free h writes
    __shared__ float sAgg[16][132];
    __shared__ float sX[16][132];
    __shared__ float sH[16][260];

    const int tid = threadIdx.x;
    const int m0 = blockIdx.x * 16;     // NN == 6250*16 exactly

    // async-stage both 16x128 tiles (16B-aligned row chunks; pad untouched)
    for (int i = tid; i < 512; i += 256) {
        int row = i >> 5;
        int c4 = (i & 31) << 2;
        int node = m0 + row;
        unsigned la = (unsigned)(uintptr_t)&sAgg[row][c4];
        const float* ga = aggN + (size_t)node * FF + c4;
        asm volatile("global_load_async_to_lds_b128 %0, %1, off"
                     :: "v"(la), "v"(ga) : "memory");
        unsigned lx = (unsigned)(uintptr_t)&sX[row][c4];
        const float* gx = x + (size_t)node * FF + c4;
        asm volatile("global_load_async_to_lds_b128 %0, %1, off"
                     :: "v"(lx), "v"(gx) : "memory");
    }
    asm volatile("s_wait_asynccnt 0" ::: "memory");
    __syncthreads();

    const int wave = tid >> 5;
    const int lane = tid & 31;
    const int rhalf = lane >> 4;    // 0: K/M low half, 1: high half
    const int l16 = lane & 15;

    for (int ct = 0; ct < 2; ++ct) {
        const int n0 = (wave * 2 + ct) * 16;
        const int n = n0 + l16;
        v8f acc = {};

        // GEMM 1: aggNorm @ W1l^T   (A: 16x4 f32 frag, B[k][n] = W1l[n][k])
#pragma unroll 4
        for (int k = 0; k < FF; k += 4) {
            int kk = k + rhalf * 2;
            v2f a;
            a.x = sAgg[l16][kk];
            a.y = sAgg[l16][kk + 1];
            float2 wv = *(const float2*)(W1l + (size_t)n * FF + kk);
            v2f b;
            b.x = wv.x;
            b.y = wv.y;
            acc = __builtin_amdgcn_wmma_f32_16x16x4_f32(false, a, false, b,
                                                        (short)0, acc, false, false);
        }
        // GEMM 2: x @ W1r^T (same accumulator)
#pragma unroll 4
        for (int k = 0; k < FF; k += 4) {
            int kk = k + rhalf * 2;
            v2f a;
            a.x = sX[l16][kk];
            a.y = sX[l16][kk + 1];
            float2 wv = *(const float2*)(W1r + (size_t)n * FF + kk);
            v2f b;
            b.x = wv.x;
            b.y = wv.y;
            acc = __builtin_amdgcn_wmma_f32_16x16x4_f32(false, a, false, b,
                                                        (short)0, acc, false, false);
        }

        // bias + ReLU; C layout: vgpr j, lanes<16 -> M=j, lanes>=16 -> M=j+8
        const float bias = b1l[n];
#pragma unroll
        for (int j = 0; j < 8; ++j) {
            int M = j + rhalf * 8;
            sH[M][n] = fmaxf(acc[j] + bias, 0.0f);
        }
    }
    __syncthreads();

    // per-row dots: wave w reduces rows w and w+8
#pragma unroll
    for (int rr = 0; rr < 2; ++rr) {
        int row = wave + rr * 8;
        float tsum = 0.0f, rsum = 0.0f;
#pragma unroll
        for (int c = lane; c < HH; c += 32) {
            float hv = sH[row][c];
            tsum += hv * W2l[c];
            rsum += hv * W2r[c];
        }
#pragma unroll
        for (int off = 16; off > 0; off >>= 1) {
            tsum += __shfl_xor(tsum, off, 32);
            rsum += __shfl_xor(rsum, off, 32);
        }
        if (lane == 0) {
            t_out[m0 + row] = tsum;
            r_out[m0 + row] = rsum;
        }
    }
}

// ---------------- kernel 6: fused conv2-mean + epilogue ----------------
__global__ __launch_bounds__(256) void final_kernel(const int* __restrict__ csr,
                                                    const int* __restrict__ offs,
                                                    const int* __restrict__ cnt,
                                                    const float* __restrict__ t,
                                                    const float* __restrict__ r,
                                                    const float* __restrict__ b2l,
                                                    const float* __restrict__ Wmu,
                                                    const float* __restrict__ bmu,
                                                    const float* __restrict__ Wlv,
                                                    const float* __restrict__ blv,
                                                    const float* __restrict__ eps,
                                                    float* __restrict__ out) {
    int i = blockIdx.x * 256 + threadIdx.x;
    if (i < NN) {
        int beg = offs[i];
        int deg = cnt[i];
        float s = 0.0f;
        for (int j = 0; j < deg; ++j) s += t[csr[beg + j]];
        float hh = s / fmaxf((float)deg, 1.0f) + b2l[0] + r[i];
        float mu = hh * Wmu[0] + bmu[0];
        float lv = hh * Wlv[0] + blv[0];
        out[i] = mu + eps[i] * expf(lv);
    }
}

extern "C" void kernel_launch(void* const* d_in, const int* in_sizes, int n_in,
                              void* d_out, int out_size, void* d_ws, size_t ws_size,
                              hipStream_t stream) {
    const float* x = (const float*)d_in[0];
    const long long* ei = (const long long*)d_in[1];
    const float* W1l = (const float*)d_in[2];
    const float* b1l = (const float*)d_in[3];
    const float* W1r = (const float*)d_in[4];
    const float* W2l = (const float*)d_in[5];
    const float* b2l = (const float*)d_in[6];
    const float* W2r = (const float*)d_in[7];
    // d_in[8..10] (Wal,bal,War): unused — softmax over size-1 axis is identity
    const float* Wmu = (const float*)d_in[11];
    const float* bmu = (const float*)d_in[12];
    const float* Wlv = (const float*)d_in[13];
    const float* blv = (const float*)d_in[14];
    const float* eps = (const float*)d_in[15];
    float* out = (float*)d_out;

    // workspace layout:
    //  floats: aggNorm[N*128] | t[N] | r[N]
    //  ints:   cnt[N] | fill[N] | offs[N] | blksum[128] | csr[E]
    float* fws = (float*)d_ws;
    float* aggNorm = fws;                              // 16B aligned at base
    float* t = fws + (size_t)NN * FF;
    float* r = t + NN;
    int* iws = (int*)(r + NN);
    int* cnt = iws;
    int* fill = cnt + NN;
    int* offs = fill + NN;
    int* blksum = offs + NN;
    int* csr = blksum + 128;

    const int nb_scan = (NN + 1023) / 1024;            // 98

    zero_int_kernel<<<256, 256, 0, stream>>>(cnt, 2 * NN);   // cnt + fill
    count_kernel<<<(NE + 255) / 256, 256, 0, stream>>>(ei, cnt);
    scan1_kernel<<<nb_scan, 256, 0, stream>>>(cnt, offs, blksum);
    scan2_kernel<<<1, 128, 0, stream>>>(blksum, nb_scan);
    scan3_kernel<<<(NN + 255) / 256, 256, 0, stream>>>(offs, blksum);
    fill_kernel<<<(NE + 255) / 256, 256, 0, stream>>>(ei, offs, fill, csr);
    aggregate_kernel<<<NN / 8, 256, 0, stream>>>(x, csr, offs, cnt, aggNorm);
    conv1_kernel<<<NN / 16, 256, 0, stream>>>(x, aggNorm, W1l, b1l, W1r,
                                              W2l, W2r, t, r);
    final_kernel<<<(NN + 255) / 256, 256, 0, stream>>>(csr, offs, cnt, t, r, b2l,
                                                       Wmu, bmu, Wlv, blv, eps, out);
}